// NonogramEnergy_59150289600875
// MI455X (gfx1250) — compile-verified
//
#include <hip/hip_runtime.h>

typedef __attribute__((ext_vector_type(2))) float v2f;
typedef __attribute__((ext_vector_type(4))) float v4f;
typedef __attribute__((ext_vector_type(8))) float v8f;

#define HAS_WMMA_F32X4 __has_builtin(__builtin_amdgcn_wmma_f32_16x16x4_f32)

static constexpr int NN = 64;      // board dim
static constexpr int NH = 16;      // hints per line
static constexpr int NB = 4096;    // batches
static constexpr int FBSZ = 12;    // fallback size

// One block (64 threads = 2 wave32) per batch.
__global__ __launch_bounds__(64) void nonogram_batch_kernel(
    const float* __restrict__ grid,   // (B,1,64,64)
    const float* __restrict__ hints,  // (B,2,64,16)
    const float* __restrict__ w_g,    // (1,64,64)
    const float* __restrict__ w_h,    // (2,64,16)
    float* __restrict__ ws)           // (B) per-batch energy contributions
{
    __shared__ float sgm[NN][65];     // sigmoid(3g) * m_row, padded -> conflict-free columns
    __shared__ float rt[NN], ct[NN];  // row/col hint targets
    __shared__ float marr[NN];        // mask as float
    __shared__ float rowsum[NN], colsum[NN];
    __shared__ float red[NN];
    __shared__ int   s_isize;

    const int b    = blockIdx.x;
    const int tid  = threadIdx.x;     // 0..63, doubles as row index
    const int lane = tid & 31;
    const int wave = tid >> 5;

    // ---------------- hints pass: thread i owns line i ----------------
    const float* h0  = hints + (size_t)b * 2 * NN * NH + tid * NH; // hints[b,0,i,:]
    const float* h1  = h0 + NN * NH;                               // hints[b,1,i,:]
    const float* wh0 = w_h + tid * NH;
    const float* wh1 = wh0 + NN * NH;
    float rsum = 0.f, csum = 0.f, dot_h = 0.f;
    #pragma unroll
    for (int h = 0; h < NH; ++h) {
        float a0 = h0[h], a1 = h1[h];
        rsum += a0; csum += a1;
        dot_h += a0 * wh0[h] + a1 * wh1[h];
    }
    rt[tid] = rsum; ct[tid] = csum;
    __syncthreads();

    if (tid == 0) {
        int last_r = -1, last_c = -1;
        for (int i = 0; i < NN; ++i) {
            if (rt[i] > 0.f) last_r = i;
            if (ct[i] > 0.f) last_c = i;
        }
        int mx = last_r > last_c ? last_r : last_c;
        s_isize = (last_r >= 0 && last_c >= 0) ? (mx + 1) : FBSZ;
    }
    __syncthreads();
    const int   isize = s_isize;
    marr[tid] = (tid < isize) ? 1.0f : 0.0f;
    __syncthreads();

    // ------- grid pass: coalesced float4 stream; dot_g, sgm, bin -------
    const float* gb = grid + (size_t)b * NN * NN;
    float dot_g = 0.f, bin = 0.f;
    #pragma unroll 4
    for (int idx4 = tid; idx4 < (NN * NN) / 4; idx4 += 64) {
        const int base = idx4 * 4;
        const int i = base >> 6, j = base & 63;
        v4f g  = *(const v4f*)(gb  + base);
        v4f wg = *(const v4f*)(w_g + base);
        const float mi = marr[i];
        float binrow = 0.f;
        #pragma unroll
        for (int c = 0; c < 4; ++c) {
            const float gv = g[c];
            dot_g += gv * wg[c];
            const float sg = 1.0f / (1.0f + __expf(-3.0f * gv));
            sgm[i][j + c] = sg * mi;               // row-masked sigmoid grid
            binrow += gv * gv * marr[j + c];
        }
        bin += binrow * mi;
    }
    __syncthreads();

#if HAS_WMMA_F32X4
    // ------- matvecs via V_WMMA_F32_16X16X4_F32 (fp32 matrix pipe) -------
    // rowsum = SGM(64x64) @ m (broadcast to 16 cols); colsum = ones(16x64) @ SGM.
    {
        const int l15   = lane & 15;
        const int halfk = (lane >> 4) << 1;       // lanes 16-31 hold K = k0+2, k0+3
        // row-sum tiles: wave w handles M-tiles {w, w+2}
        for (int Mt = wave; Mt < 4; Mt += 2) {
            v8f acc = {};
            const int mrow = Mt * 16 + l15;
            for (int k0 = 0; k0 < NN; k0 += 4) {
                const int kk = k0 + halfk;
                v2f a;  a.x  = sgm[mrow][kk]; a.y  = sgm[mrow][kk + 1];  // A = SGM tile
                v2f bb; bb.x = marr[kk];      bb.y = marr[kk + 1];        // B[k][n] = m_k
                acc = __builtin_amdgcn_wmma_f32_16x16x4_f32(
                          false, a, false, bb, (short)0, acc, false, false);
            }
            if (l15 == 0) {                        // C: all 16 cols identical; rows r+8*(lane>=16)
                const int roff = Mt * 16 + ((lane >> 4) << 3);
                #pragma unroll
                for (int r = 0; r < 8; ++r) rowsum[roff + r] = acc[r];
            }
        }
        // col-sum tiles: wave w handles N-tiles {w, w+2}
        for (int Nt = wave; Nt < 4; Nt += 2) {
            v8f acc = {};
            const int col = Nt * 16 + l15;
            for (int k0 = 0; k0 < NN; k0 += 4) {
                const int kk = k0 + halfk;
                v2f a;  a.x  = 1.0f;            a.y  = 1.0f;              // A = ones
                v2f bb; bb.x = sgm[kk][col];    bb.y = sgm[kk + 1][col];  // B = SGM tile
                acc = __builtin_amdgcn_wmma_f32_16x16x4_f32(
                          false, a, false, bb, (short)0, acc, false, false);
            }
            if (lane < 16) colsum[col] = acc[0];   // all rows of C identical
        }
    }
#else
    // scalar fallback (histogram will show wmma=0 -> iterate on builtin name)
    {
        float rs = 0.f, cs = 0.f;
        for (int k = 0; k < NN; ++k) {
            rs += sgm[tid][k] * marr[k];
            cs += sgm[k][tid];
        }
        rowsum[tid] = rs; colsum[tid] = cs;
    }
#endif
    __syncthreads();

    // ---------------- per-line errors + block combine ----------------
    const float szf = (float)isize;
    const float mi  = marr[tid];
    float re = rowsum[tid] - rt[tid]; re = re * re * mi;  // rowsum already row-masked; ok since *mi
    float ce = colsum[tid] - ct[tid]; ce = ce * ce * mi;
    red[tid] = dot_g + dot_h
             + 10.0f * (re + ce) / szf
             + 0.1f  * bin / (szf * szf);
    __syncthreads();
    if (tid == 0) {
        float s = 0.f;
        for (int i = 0; i < NN; ++i) s += red[i];  // fixed order -> deterministic
        ws[b] = s;
    }
}

// Deterministic final reduction of 4096 per-batch partials.
__global__ __launch_bounds__(256) void nonogram_reduce_kernel(
    const float* __restrict__ ws, float* __restrict__ out)
{
    __shared__ float red[256];
    const int t = threadIdx.x;
    float s = 0.f;
    for (int i = t; i < NB; i += 256) s += ws[i];
    red[t] = s;
    __syncthreads();
    for (int stride = 128; stride > 0; stride >>= 1) {
        if (t < stride) red[t] += red[t + stride];
        __syncthreads();
    }
    if (t == 0) out[0] = red[0] * (1.0f / (float)NB);
}

extern "C" void kernel_launch(void* const* d_in, const int* in_sizes, int n_in,
                              void* d_out, int out_size, void* d_ws, size_t ws_size,
                              hipStream_t stream) {
    const float* grid  = (const float*)d_in[0];
    const float* hints = (const float*)d_in[1];
    const float* w_g   = (const float*)d_in[2];
    const float* w_h   = (const float*)d_in[3];
    float* ws = (float*)d_ws;   // needs 4096 floats = 16 KB scratch

    nonogram_batch_kernel<<<NB, 64, 0, stream>>>(grid, hints, w_g, w_h, ws);
    nonogram_reduce_kernel<<<1, 256, 0, stream>>>(ws, (float*)d_out);
}